// GlobalPointerModel_8383776162434
// MI455X (gfx1250) — compile-verified
//
#include <hip/hip_runtime.h>
#include <hip/hip_bf16.h>
#include <math.h>

// ---------------------------------------------------------------------------
// GlobalPointer model forward (loss + logits) for MI455X / gfx1250.
//
//   proj  = X(8192x768) @ W(768x1024) + b     f32 WMMA 16x16x4,
//                                             A panel staged via TDM -> LDS
//   q,k   = RoPE(proj halves)                 fused epilogue (shfl_xor pair)
//   logit = q @ k^T per (b,e), mask, tril, /8 f32 WMMA 16x16x4
//   loss  = multilabel CCE, streaming logsumexp, fixed-order (deterministic)
//
// d_out = [loss(1)] ++ [logits(16*8*512*512)]  (fp32)
// d_ws  = qA (16MB) ++ kT (16MB) ++ partials (256KB)
// ---------------------------------------------------------------------------

#define ENT   8
#define INNER 64
#define BDIM  16
#define SDIM  512
#define HDIM  768
#define NOUT  1024
#define BIGC  1000000000000.0f

typedef float        v2f __attribute__((ext_vector_type(2)));
typedef float        v8f __attribute__((ext_vector_type(8)));
typedef unsigned int v4u __attribute__((ext_vector_type(4)));
typedef int          v4i __attribute__((ext_vector_type(4)));
typedef int          v8i __attribute__((ext_vector_type(8)));

#define WMMA_F32_16x16x4(a, b, c) \
    __builtin_amdgcn_wmma_f32_16x16x4_f32(false, (a), false, (b), (short)0, (c), false, false)

// --- streaming logsumexp helpers (deterministic fixed-order combines) ------
__device__ __forceinline__ void lse_add(float& m, float& s, float x) {
    if (x > m) { s = s * __expf(m - x) + 1.0f; m = x; }
    else       { s += __expf(x - m); }
}
__device__ __forceinline__ void lse_combine(float& m1, float& s1, float m2, float s2) {
    if (s2 == 0.0f) return;
    if (m2 > m1) { s1 = s1 * __expf(m1 - m2) + s2; m1 = m2; }
    else         { s1 += s2 * __expf(m2 - m1); }
}

// ---------------------------------------------------------------------------
// Kernel 1: projection GEMM + bias + RoPE.
// Block = 256 thr (8 waves) = one 16-row M panel x 8 consecutive 16-col N
// tiles. The shared 16x768 A panel is DMA'd into LDS once by the TDM
// (pad: +2 DWORDs every 256 -> padded row stride 774 DWORDs, conflict-free
// 16-lane reads, 8B aligned). Inner loop: ds_load_b64 (A) + 2x b32 (B) + WMMA.
// ---------------------------------------------------------------------------
#define A_PANEL_DWORDS (16 * HDIM + 2 * ((16 * HDIM) / 256))   // 12384
#define A_ROW_STRIDE   774                                     // 768 + 2*3

__global__ __launch_bounds__(256)
void proj_rope_kernel(const float* __restrict__ X,     // [8192][768]
                      const float* __restrict__ W,     // [768][1024]
                      const float* __restrict__ bias,  // [1024]
                      const int*   __restrict__ bbox,  // [8192][4]
                      float* __restrict__ qA,          // [128][512][64]
                      float* __restrict__ kT)          // [128][64][512]
{
    __shared__ float As[A_PANEL_DWORDS];

    const int lane = threadIdx.x & 31;
    const int wave = threadIdx.x >> 5;

    const int tileM = blockIdx.x >> 3;                 // block-uniform M tile
    const int m0    = tileM * 16;
    const int tileN = ((blockIdx.x & 7) * 8 + wave);   // 8 N tiles per block
    const int n0    = tileN * 16;

    // ---- TDM: stage A panel (16 x 768 f32 @ row m0) into LDS --------------
    if (wave == 0) {
        const unsigned long long gaddr =
            (unsigned long long)(const void*)(X + (size_t)m0 * HDIM);
        const unsigned int lds_addr = (unsigned int)(size_t)(void*)&As[0];

        // D# group 0: count=1 | lds_addr | global_addr[56:0] | type=2
        v4u g0 = { 1u,
                   lds_addr,
                   (unsigned int)(gaddr & 0xFFFFFFFFull),
                   (unsigned int)((gaddr >> 32) & 0x01FFFFFFull) | 0x80000000u };

        // D# group 1:
        //  w0: wg_mask=0 | data_size=2(4B) | pad_enable | pad_interval=7(256DW)
        //      | pad_amount=1(2DW)
        //  tensor_dim0=768, tensor_dim1=16, tile_dim0=768, tile_dim1=16,
        //  tensor_dim0_stride=768
        v8i g1 = { (int)((2u << 16) | (1u << 20) | (7u << 22) | (1u << 25)),
                   (int)(768u << 16),          // tensor_dim0[15:0]
                   (int)(16u << 16),           // tensor_dim0[31:16]=0 | tensor_dim1[15:0]
                   (int)(768u << 16),          // tensor_dim1[31:16]=0 | tile_dim0
                   16,                         // tile_dim1 | tile_dim2=0
                   768,                        // tensor_dim0_stride[31:0]
                   0, 0 };

        v4i gz4 = { 0, 0, 0, 0 };              // groups 2/3 unused (2D tensor)
        v8i gz8 = { 0, 0, 0, 0, 0, 0, 0, 0 };  // extra group (clang-23 6-arg form)
        __builtin_amdgcn_tensor_load_to_lds(g0, g1, gz4, gz4, gz8, 0);
        __builtin_amdgcn_s_wait_tensorcnt(0);
    }
    __syncthreads();

    const int lm = lane & 15;                  // M (A) / N (B) index
    const int kh = lane >> 4;                  // K half select

    const float* Ap   = As + lm * A_ROW_STRIDE;
    const int    ncol = n0 + lm;

    v8f acc = {0.f, 0.f, 0.f, 0.f, 0.f, 0.f, 0.f, 0.f};

    for (int kk = 0; kk < HDIM; kk += 4) {
        // A frag from LDS: lane holds M=lm, K = kk + kh*2 + {0,1}
        v2f afrag = *(const v2f*)(Ap + kk + kh * 2 + 2 * (kk >> 8));
        // B frag from global: lane holds N=ncol, K = kk + kh*2 + {0,1}
        v2f bfrag;
        bfrag.x = W[(size_t)(kk + kh * 2    ) * NOUT + ncol];
        bfrag.y = W[(size_t)(kk + kh * 2 + 1) * NOUT + ncol];
        acc = WMMA_F32_16x16x4(afrag, bfrag, acc);
    }

    const float bv = bias[ncol];

    // RoPE parameters for this column
    const int e    = ncol >> 7;
    const int w    = ncol & 127;
    const int is_k = (w >= INNER);
    const int d    = is_k ? (w - INNER) : w;   // 0..63
    const int j    = d & 31;
    const float inv_freq = __powf(10000.0f, -(float)(j >> 1) * (1.0f / 16.0f));

    #pragma unroll
    for (int r = 0; r < 8; ++r) {
        const int m = m0 + r + kh * 8;         // token index
        const int b = m >> 9;
        const int s = m & 511;

        const int px  = bbox[m * 4 + 0];
        const int py  = bbox[m * 4 + 1];
        const int pos = (d < 32) ? px : py;

        const float ang = (float)pos * inv_freq;
        float cs, sn;
        __sincosf(ang, &sn, &cs);

        const float v  = acc[r] + bv;
        const float pv = __shfl_xor(v, 1, 32); // rotate_half partner
        const float rh = (d & 1) ? pv : -pv;
        const float out = v * cs + rh * sn;

        const int be = b * ENT + e;
        if (!is_k) qA[((size_t)be * SDIM + s) * INNER + d] = out;
        else       kT[((size_t)be * INNER + d) * SDIM + s] = out;
    }
}

// ---------------------------------------------------------------------------
// Kernel 2: logits = qA @ kT (per b,e) + masks + scale; fused loss partials.
// ---------------------------------------------------------------------------
__global__ __launch_bounds__(256)
void logits_loss_kernel(const float* __restrict__ qA,      // [128][512][64]
                        const float* __restrict__ kT,      // [128][64][512]
                        const int*   __restrict__ amask,   // [16][512]
                        const int*   __restrict__ labels,  // [128][512][512]
                        float* __restrict__ out,           // loss @0, logits @1
                        float* __restrict__ partials)      // [128][128][4]
{
    __shared__ float red[8][4];

    const int lane = threadIdx.x & 31;
    const int wave = threadIdx.x >> 5;
    const int be   = blockIdx.y;
    const int b    = be >> 3;

    const int tile  = blockIdx.x * 8 + wave;
    const int tileM = tile >> 5;
    const int tileN = tile & 31;
    const int m0 = tileM * 16;
    const int n0 = tileN * 16;

    const int lm = lane & 15;
    const int kh = lane >> 4;

    const float* Arow  = qA + ((size_t)be * SDIM + (m0 + lm)) * INNER;
    const float* Bbase = kT + (size_t)be * INNER * SDIM;
    const int    ncol  = n0 + lm;

    v8f acc = {0.f, 0.f, 0.f, 0.f, 0.f, 0.f, 0.f, 0.f};

    #pragma unroll
    for (int dd = 0; dd < INNER; dd += 4) {
        v2f afrag = *(const v2f*)(Arow + dd + kh * 2);
        v2f bfrag;
        bfrag.x = Bbase[(size_t)(dd + kh * 2    ) * SDIM + ncol];
        bfrag.y = Bbase[(size_t)(dd + kh * 2 + 1) * SDIM + ncol];
        acc = WMMA_F32_16x16x4(afrag, bfrag, acc);
    }

    __builtin_prefetch(labels + ((size_t)be * SDIM + m0 + kh * 8) * SDIM + ncol, 0, 0);

    const float pad = (float)amask[b * SDIM + ncol];

    float mn = -__builtin_inff(), sn = 0.0f;
    float mp = -__builtin_inff(), sp = 0.0f;

    #pragma unroll
    for (int r = 0; r < 8; ++r) {
        const int m = m0 + r + kh * 8;
        const int n = ncol;

        float lg = acc[r];
        lg = lg * pad - (1.0f - pad) * BIGC;
        if (m > n) lg -= BIGC;
        lg *= 0.125f;

        out[1 + ((size_t)be * SDIM + m) * SDIM + n] = lg;

        const float y  = (float)labels[((size_t)be * SDIM + m) * SDIM + n];
        const float yp = (1.0f - 2.0f * y) * lg;
        lse_add(mn, sn, yp - y * BIGC);
        lse_add(mp, sp, yp - (1.0f - y) * BIGC);
    }

    #pragma unroll
    for (int off = 16; off > 0; off >>= 1) {
        float m2, s2;
        m2 = __shfl_down(mn, off, 32); s2 = __shfl_down(sn, off, 32);
        lse_combine(mn, sn, m2, s2);
        m2 = __shfl_down(mp, off, 32); s2 = __shfl_down(sp, off, 32);
        lse_combine(mp, sp, m2, s2);
    }
    if (lane == 0) {
        red[wave][0] = mn; red[wave][1] = sn;
        red[wave][2] = mp; red[wave][3] = sp;
    }
    __syncthreads();

    if (threadIdx.x == 0) {
        float Mn = red[0][0], Sn = red[0][1], Mp = red[0][2], Sp = red[0][3];
        #pragma unroll
        for (int i = 1; i < 8; ++i) {
            lse_combine(Mn, Sn, red[i][0], red[i][1]);
            lse_combine(Mp, Sp, red[i][2], red[i][3]);
        }
        float* p = partials + ((size_t)be * 128 + blockIdx.x) * 4;
        p[0] = Mn; p[1] = Sn; p[2] = Mp; p[3] = Sp;
    }
}

// ---------------------------------------------------------------------------
// Kernel 3: combine partials per (b,e) row in fixed order -> scalar loss.
// ---------------------------------------------------------------------------
__global__ __launch_bounds__(128)
void loss_reduce_kernel(const float* __restrict__ partials, float* __restrict__ out)
{
    __shared__ float rowloss[128];
    const int be = threadIdx.x;

    float mn = -__builtin_inff(), sn = 0.0f;
    float mp = -__builtin_inff(), sp = 0.0f;
    for (int i = 0; i < 128; ++i) {
        const float* p = partials + ((size_t)be * 128 + i) * 4;
        lse_combine(mn, sn, p[0], p[1]);
        lse_combine(mp, sp, p[2], p[3]);
    }
    lse_combine(mn, sn, 0.0f, 1.0f);   // appended zero element
    lse_combine(mp, sp, 0.0f, 1.0f);

    rowloss[be] = (mn + __logf(sn)) + (mp + __logf(sp));
    __syncthreads();

    if (be == 0) {
        float acc = 0.0f;
        for (int i = 0; i < 128; ++i) acc += rowloss[i];
        out[0] = acc * (1.0f / 128.0f);
    }
}

// ---------------------------------------------------------------------------
extern "C" void kernel_launch(void* const* d_in, const int* in_sizes, int n_in,
                              void* d_out, int out_size, void* d_ws, size_t ws_size,
                              hipStream_t stream) {
    const float* X      = (const float*)d_in[0];
    const float* W      = (const float*)d_in[1];
    const float* bias   = (const float*)d_in[2];
    const int*   amask  = (const int*)  d_in[3];
    const int*   bbox   = (const int*)  d_in[4];
    const int*   labels = (const int*)  d_in[5];

    float* out = (float*)d_out;

    const size_t QA_ELEMS = (size_t)BDIM * ENT * SDIM * INNER;  // 4,194,304
    float* qA       = (float*)d_ws;
    float* kT       = qA + QA_ELEMS;
    float* partials = kT + QA_ELEMS;

    proj_rope_kernel<<<dim3(4096), dim3(256), 0, stream>>>(X, W, bias, bbox, qA, kT);

    logits_loss_kernel<<<dim3(128, 128), dim3(256), 0, stream>>>(
        qA, kT, amask, labels, out, partials);

    loss_reduce_kernel<<<dim3(1), dim3(128), 0, stream>>>(partials, out);
}